// PlenoxelModel_57191784513782
// MI455X (gfx1250) — compile-verified
//
#include <hip/hip_runtime.h>

typedef __attribute__((ext_vector_type(2))) float v2f;
typedef __attribute__((ext_vector_type(8))) float v8f;

#define RGRID 192
#define FDIM  28

// SH constants
#define C0f  0.28209479177387814f
#define C1f  0.4886025119029199f
#define C2f  0.5462742152960396f
#define C20f 0.15769578262626002f
#define C22f 0.2731371076480198f

__global__ __launch_bounds__(256) void plenoxel_fused(
    const float* __restrict__ points,
    const float* __restrict__ voxels,
    float* __restrict__ out,
    int N)
{
    const int lane = threadIdx.x & 31;
    const int wave = threadIdx.x >> 5;
    const int pin  = lane & 15;           // point-in-wave; lanes l and l+16 mirror
    long long g = (long long)blockIdx.x * 128 + wave * 16 + pin;
    const bool valid = g < (long long)N;
    const long long gc = valid ? g : (long long)(N - 1);   // keep EXEC full for WMMA

    // ---- load point ----
    const float px = points[gc * 3 + 0];
    const float py = points[gc * 3 + 1];
    const float pz = points[gc * 3 + 2];

    // ---- trilinear setup (pts_norm = p/4) ----
    const float sc = 0.5f * (float)(RGRID - 1);
    float cx = fminf(fmaxf((px * 0.25f + 1.0f) * sc, 0.0f), (float)(RGRID - 1));
    float cy = fminf(fmaxf((py * 0.25f + 1.0f) * sc, 0.0f), (float)(RGRID - 1));
    float cz = fminf(fmaxf((pz * 0.25f + 1.0f) * sc, 0.0f), (float)(RGRID - 1));
    float flx = floorf(cx), fly = floorf(cy), flz = floorf(cz);
    float fx = cx - flx, fy = cy - fly, fz = cz - flz;
    int ix0 = (int)flx, iy0 = (int)fly, iz0 = (int)flz;
    int ix1 = min(ix0 + 1, RGRID - 1);
    int iy1 = min(iy0 + 1, RGRID - 1);
    int iz1 = min(iz0 + 1, RGRID - 1);

    const float wx0 = 1.0f - fx, wx1 = fx;
    const float wy0 = 1.0f - fy, wy1 = fy;
    const float wz0 = 1.0f - fz, wz1 = fz;
    const float w000 = wz0 * wy0 * wx0, w001 = wz0 * wy0 * wx1;
    const float w010 = wz0 * wy1 * wx0, w011 = wz0 * wy1 * wx1;
    const float w100 = wz1 * wy0 * wx0, w101 = wz1 * wy0 * wx1;
    const float w110 = wz1 * wy1 * wx0, w111 = wz1 * wy1 * wx1;

    // corner element offsets (max 192^3*28 ≈ 1.98e8 < 2^31)
    const int o000 = (((iz0 * RGRID) + iy0) * RGRID + ix0) * FDIM;
    const int o001 = (((iz0 * RGRID) + iy0) * RGRID + ix1) * FDIM;
    const int o010 = (((iz0 * RGRID) + iy1) * RGRID + ix0) * FDIM;
    const int o011 = (((iz0 * RGRID) + iy1) * RGRID + ix1) * FDIM;
    const int o100 = (((iz1 * RGRID) + iy0) * RGRID + ix0) * FDIM;
    const int o101 = (((iz1 * RGRID) + iy0) * RGRID + ix1) * FDIM;
    const int o110 = (((iz1 * RGRID) + iy1) * RGRID + ix0) * FDIM;
    const int o111 = (((iz1 * RGRID) + iy1) * RGRID + ix1) * FDIM;

    // ---- gather + trilerp, 7 chunks of float4 (28 features) ----
    float feats[FDIM];
#pragma unroll
    for (int ch = 0; ch < 7; ++ch) {
        const int fo = ch * 4;
        float4 a0 = *(const float4*)(voxels + o000 + fo);
        float4 a1 = *(const float4*)(voxels + o001 + fo);
        float4 a2 = *(const float4*)(voxels + o010 + fo);
        float4 a3 = *(const float4*)(voxels + o011 + fo);
        float4 a4 = *(const float4*)(voxels + o100 + fo);
        float4 a5 = *(const float4*)(voxels + o101 + fo);
        float4 a6 = *(const float4*)(voxels + o110 + fo);
        float4 a7 = *(const float4*)(voxels + o111 + fo);
        feats[fo + 0] = w000*a0.x + w001*a1.x + w010*a2.x + w011*a3.x
                      + w100*a4.x + w101*a5.x + w110*a6.x + w111*a7.x;
        feats[fo + 1] = w000*a0.y + w001*a1.y + w010*a2.y + w011*a3.y
                      + w100*a4.y + w101*a5.y + w110*a6.y + w111*a7.y;
        feats[fo + 2] = w000*a0.z + w001*a1.z + w010*a2.z + w011*a3.z
                      + w100*a4.z + w101*a5.z + w110*a6.z + w111*a7.z;
        feats[fo + 3] = w000*a0.w + w001*a1.w + w010*a2.w + w011*a3.w
                      + w100*a4.w + w101*a5.w + w110*a6.w + w111*a7.w;
    }

    // ---- SH basis (K padded to 12 with zeros) ----
    const float rn = 1.0f / sqrtf(px*px + py*py + pz*pz);
    const float dx = px * rn, dy = py * rn, dz = pz * rn;
    float bb[12];
    bb[0] = C0f;
    bb[1] = C1f * dy;
    bb[2] = C1f * dz;
    bb[3] = C1f * dx;
    bb[4] = C2f * dx * dy;
    bb[5] = C2f * dy * dz;
    bb[6] = C20f * (3.0f * dz * dz - 1.0f);
    bb[7] = C2f * dx * dz;
    bb[8] = C22f * (dx * dx - dy * dy);
    bb[9] = 0.0f; bb[10] = 0.0f; bb[11] = 0.0f;

    const bool lo = (lane < 16);

    // ---- rgb[c] = sigmoid( Σ_k sh[c,k]*bb[k] ) via chained f32 WMMA (diagonal) ----
    float rgb[3];
#pragma unroll
    for (int c = 0; c < 3; ++c) {
        float shc[12];
#pragma unroll
        for (int k = 0; k < 9; ++k) shc[k] = feats[1 + c * 9 + k];
        shc[9] = 0.0f; shc[10] = 0.0f; shc[11] = 0.0f;

        v8f acc = {0.f, 0.f, 0.f, 0.f, 0.f, 0.f, 0.f, 0.f};
#pragma unroll
        for (int kc = 0; kc < 12; kc += 4) {
            // A (16x4 f32): lanes 0-15 -> K=kc,kc+1 ; lanes 16-31 -> K=kc+2,kc+3
            v2f a, b;
            a.x = lo ? bb[kc]     : bb[kc + 2];
            a.y = lo ? bb[kc + 1] : bb[kc + 3];
            // B (4x16 f32): same half-wave K split, N = lane&15 (mirrored point)
            b.x = lo ? shc[kc]     : shc[kc + 2];
            b.y = lo ? shc[kc + 1] : shc[kc + 3];
            acc = __builtin_amdgcn_wmma_f32_16x16x4_f32(
                false, a, false, b, (short)0, acc, false, false);
        }
        // diagonal D[p][p]: owner lanes (l<8: vgpr l) / (l>=24: vgpr l-24) -> vgpr lane&7
        const int r7 = lane & 7;
        float v0 = (r7 & 1) ? acc[1] : acc[0];
        float v1 = (r7 & 1) ? acc[3] : acc[2];
        float v2 = (r7 & 1) ? acc[5] : acc[4];
        float v3 = (r7 & 1) ? acc[7] : acc[6];
        float u0 = (r7 & 2) ? v1 : v0;
        float u1 = (r7 & 2) ? v3 : v2;
        float dsum = (r7 & 4) ? u1 : u0;
        rgb[c] = 1.0f / (1.0f + expf(-dsum));
    }

    // ---- sigma = softplus(density), numerically stable ----
    const float den = feats[0];
    const float sigma = fmaxf(den, 0.0f) + log1pf(expf(-fabsf(den)));

    // ---- owner lanes write (one lane per point) ----
    if (valid && ((lane < 8) || (lane >= 24))) {
        out[g * 3 + 0] = rgb[0];
        out[g * 3 + 1] = rgb[1];
        out[g * 3 + 2] = rgb[2];
        out[(long long)3 * N + g] = sigma;
    }
}

extern "C" void kernel_launch(void* const* d_in, const int* in_sizes, int n_in,
                              void* d_out, int out_size, void* d_ws, size_t ws_size,
                              hipStream_t stream)
{
    (void)n_in; (void)out_size; (void)d_ws; (void)ws_size;
    const float* points = (const float*)d_in[0];
    const float* voxels = (const float*)d_in[1];
    float* out = (float*)d_out;
    const int N = in_sizes[0] / 3;             // points is N x 3
    const int blocks = (N + 127) / 128;        // 16 points/wave, 8 waves/block
    hipLaunchKernelGGL(plenoxel_fused, dim3(blocks), dim3(256), 0, stream,
                       points, voxels, out, N);
}